// Dumplicate_Removal_88819923681511
// MI455X (gfx1250) — compile-verified
//
#include <hip/hip_runtime.h>
#include <math.h>
#include <stdint.h>

typedef __attribute__((ext_vector_type(2))) float v2f;
typedef __attribute__((ext_vector_type(8))) float v8f;

#define N_OBJ   256
#define DIM_MM  1024
#define DIM_HO  4096
#define DIM_KEY 512

// ---------------------------------------------------------------------------
// CDNA5 async global->LDS copies (ASYNCcnt-tracked, no VGPR bounce).
// Per ISA: flat shared-memory addresses keep the wave-relative LDS byte
// offset in addr[31:0], so the low 32 bits of a generic __shared__ pointer
// are a valid per-lane LDS address (VDST) for these instructions.
// ---------------------------------------------------------------------------
__device__ __forceinline__ void async_load_b32(void* lds_ptr, const void* gptr) {
    uint32_t lds = (uint32_t)(uintptr_t)lds_ptr;
    uint64_t ga  = (uint64_t)(uintptr_t)gptr;
    asm volatile("global_load_async_to_lds_b32 %0, %1, off"
                 :: "v"(lds), "v"(ga) : "memory");
}
__device__ __forceinline__ void async_load_b128(void* lds_ptr, const void* gptr) {
    uint32_t lds = (uint32_t)(uintptr_t)lds_ptr;
    uint64_t ga  = (uint64_t)(uintptr_t)gptr;
    asm volatile("global_load_async_to_lds_b128 %0, %1, off"
                 :: "v"(lds), "v"(ga) : "memory");
}
__device__ __forceinline__ void wait_async_all() {
    asm volatile("s_wait_asynccnt 0x0" ::: "memory");
}

// ---------------------------------------------------------------------------
// rank[pos] = index of pos-th largest element of p (stable, matches argsort(-p))
// ---------------------------------------------------------------------------
__global__ __launch_bounds__(256)
void rank_kernel(const float* __restrict__ p, int* __restrict__ rank) {
    __shared__ float sp[N_OBJ];
    int i = threadIdx.x;
    sp[i] = p[i];
    __syncthreads();
    float pi = sp[i];
    int cnt = 0;
    for (int m = 0; m < N_OBJ; ++m) {
        float pm = sp[m];
        cnt += (pm > pi) || (pm == pi && m < i);
    }
    rank[cnt] = i;
}

// ---------------------------------------------------------------------------
// Generic fp32 WMMA GEMM:
//   D[m][n] = act( scale * sum_k A[m][k]*B[k][n] + bias[n] + add_rows[row_idx[m]][n] )
// with B[k][n] = Bp[n*ldbn + k*ldbk]   (covers A@W^T and A@B)
// Block: 256 threads = 8 waves; tile 64(M) x 64(N); K staged in LDS chunks
// of 32 via async global->LDS. Each wave owns two 16x16 subtiles sharing one
// A fragment (16 v_wmma per barrier pair).
// ---------------------------------------------------------------------------
#define BM 64
#define BN 64
#define BK 32
#define APAD 4   // sA row = BK+4 = 36 floats = 144B -> 16B-aligned rows for b128
#define BPAD 4

__global__ __launch_bounds__(256)
void gemm_wmma_f32(const float* __restrict__ A, const float* __restrict__ Bp,
                   float* __restrict__ D,
                   const float* __restrict__ bias,
                   const float* __restrict__ add_rows,
                   const int* __restrict__ row_idx,
                   int M, int N, int K, int ldbn, int ldbk,
                   float scale, int act)
{
    __shared__ float sA[BM][BK + APAD];   // 64 x 36 floats
    __shared__ float sB[BK][BN + BPAD];   // 32 x 68 floats

    const int tid  = threadIdx.x;
    const int wave = tid >> 5;
    const int lane = tid & 31;
    const int half = lane >> 4;     // 0: lanes 0-15, 1: lanes 16-31
    const int l16  = lane & 15;
    const int wm   = (wave & 3) * 16;   // wave M-offset inside block tile
    const int wn0  = (wave >> 2) * 32;  // wave N-offset (two 16-wide subtiles)
    const int m0   = blockIdx.y * BM;
    const int n0   = blockIdx.x * BN;

    v8f acc0 = {0.f, 0.f, 0.f, 0.f, 0.f, 0.f, 0.f, 0.f};
    v8f acc1 = {0.f, 0.f, 0.f, 0.f, 0.f, 0.f, 0.f, 0.f};

    for (int kk = 0; kk < K; kk += BK) {
        // --- A tile: 64 x 32 floats = 512 float4 chunks, 2 per thread (b128) ---
#pragma unroll
        for (int u = 0; u < 2; ++u) {
            int c    = tid + u * 256;     // 0..511
            int row  = c >> 3;            // 0..63
            int col4 = (c & 7) * 4;       // 0,4,...,28
            async_load_b128(&sA[row][col4],
                            A + (size_t)(m0 + row) * K + kk + col4);
        }
        // --- B tile: 32 x 64 floats, 8 scalars per thread (b32, generic strides) ---
#pragma unroll
        for (int u = 0; u < 8; ++u) {
            int idx = tid + u * 256;      // 0..2047
            int kB  = idx >> 6;           // 0..31
            int nB  = idx & 63;           // 0..63
            async_load_b32(&sB[kB][nB],
                           Bp + (size_t)(n0 + nB) * ldbn + (size_t)(kk + kB) * ldbk);
        }
        wait_async_all();
        __syncthreads();

#pragma unroll
        for (int k0 = 0; k0 < BK; k0 += 4) {
            v2f a, b0, b1;
            // A 16x4 fp32 layout: lanes 0-15 -> K{0,1}, lanes 16-31 -> K{2,3}, M = lane&15
            a.x  = sA[wm + l16][k0 + 2 * half + 0];
            a.y  = sA[wm + l16][k0 + 2 * half + 1];
            // B 4x16 fp32 layout mirrored over N
            b0.x = sB[k0 + 2 * half + 0][wn0 + l16];
            b0.y = sB[k0 + 2 * half + 1][wn0 + l16];
            b1.x = sB[k0 + 2 * half + 0][wn0 + 16 + l16];
            b1.y = sB[k0 + 2 * half + 1][wn0 + 16 + l16];
            acc0 = __builtin_amdgcn_wmma_f32_16x16x4_f32(
                false, a, false, b0, (short)0, acc0, false, false);
            acc1 = __builtin_amdgcn_wmma_f32_16x16x4_f32(
                false, a, false, b1, (short)0, acc1, false, false);
        }
        __syncthreads();
    }

    // C/D layout: VGPR r -> M = r + 8*half, N = lane&15
#pragma unroll
    for (int r = 0; r < 8; ++r) {
        int m  = m0 + wm + r + 8 * half;
        int n1 = n0 + wn0 + l16;
        int n2 = n1 + 16;
        float v0 = acc0[r] * scale;
        float v1 = acc1[r] * scale;
        if (bias)     { v0 += bias[n1];  v1 += bias[n2]; }
        if (add_rows) {
            const float* arow = add_rows + (size_t)row_idx[m] * N;
            v0 += arow[n1];  v1 += arow[n2];
        }
        if (act) { v0 = fmaxf(v0, 0.f); v1 = fmaxf(v1, 0.f); }
        D[(size_t)m * N + n1] = v0;
        D[(size_t)m * N + n2] = v1;
    }
}

// ---------------------------------------------------------------------------
// Geometry weight: gw[i][j] = relu( pe(i,j) . Wg + bg )
// pe(i,j) = concat over c in 0..3 of [sin(g_c * s_d), cos(g_c * s_d)] d=0..63
// with s_d = 100 * 1000^(-8d/512), Wg layout: sin at c*128+d, cos at c*128+64+d
// ---------------------------------------------------------------------------
__global__ __launch_bounds__(256)
void geometry_kernel(const float* __restrict__ rois,
                     const float* __restrict__ Wg,
                     const float* __restrict__ bg,
                     float* __restrict__ gw)
{
    __shared__ float sWg[DIM_KEY];
    __shared__ float sS[64];
    int i = blockIdx.x;
    int j = threadIdx.x;
    sWg[j]       = Wg[j];
    sWg[j + 256] = Wg[j + 256];
    if (j < 64) {
        sS[j] = 100.f * expf(-(8.f * (float)j / 512.f) * logf(1000.f));
    }
    __syncthreads();

    float xi1 = rois[i * 4 + 0], yi1 = rois[i * 4 + 1];
    float xi2 = rois[i * 4 + 2], yi2 = rois[i * 4 + 3];
    float xj1 = rois[j * 4 + 0], yj1 = rois[j * 4 + 1];
    float xj2 = rois[j * 4 + 2], yj2 = rois[j * 4 + 3];
    float wi = xi2 - xi1 + 1e-10f, hi = yi2 - yi1 + 1e-10f;
    float wj = xj2 - xj1 + 1e-10f, hj = yj2 - yj1 + 1e-10f;
    float cxi = (xi1 + xi2) * 0.5f, cyi = (yi1 + yi2) * 0.5f;
    float cxj = (xj1 + xj2) * 0.5f, cyj = (yj1 + yj2) * 0.5f;

    float f[4];
    f[0] = fabsf(cxi - cxj) / wi;
    f[1] = fabsf(cyi - cyj) / hi;
    f[2] = wi / wj;
    f[3] = hi / hj;

    float acc = 0.f;
#pragma unroll
    for (int c = 0; c < 4; ++c) {
        float gl = (f[c] == 0.f) ? 0.f : logf(f[c]);
        const float* wgp = sWg + c * 128;
        for (int d = 0; d < 64; ++d) {
            float ang = gl * sS[d];
            float s, co;
            sincosf(ang, &s, &co);
            acc += s * wgp[d] + co * wgp[64 + d];
        }
    }
    float r = acc + bg[0];
    gw[i * N_OBJ + j] = r > 0.f ? r : 0.f;
}

// ---------------------------------------------------------------------------
// att[i][j] = exp(vw[i][j]) * gw[i][j], zero diagonal, row-normalized
// ---------------------------------------------------------------------------
__global__ __launch_bounds__(256)
void att_kernel(const float* __restrict__ vw, const float* __restrict__ gw,
                float* __restrict__ att)
{
    __shared__ float red[256];
    int i = blockIdx.x;
    int j = threadIdx.x;
    float a = expf(vw[i * N_OBJ + j]) * gw[i * N_OBJ + j];
    if (j == i) a = 0.f;
    red[j] = a;
    __syncthreads();
    for (int s = 128; s > 0; s >>= 1) {
        if (j < s) red[j] += red[j + s];
        __syncthreads();
    }
    float denom = red[0] + 1e-10f;
    att[i * N_OBJ + j] = a / denom;
}

// ---------------------------------------------------------------------------
// out[i] = sigmoid( dot(featR[i,:], Wr) + br )   (feat already ReLU'd)
// ---------------------------------------------------------------------------
__global__ __launch_bounds__(256)
void final_kernel(const float* __restrict__ feat, const float* __restrict__ Wr,
                  const float* __restrict__ br, float* __restrict__ out)
{
    __shared__ float red[256];
    int i = blockIdx.x;
    int t = threadIdx.x;
    float acc = 0.f;
    for (int c = t; c < DIM_MM; c += 256)
        acc += feat[(size_t)i * DIM_MM + c] * Wr[c];
    red[t] = acc;
    __syncthreads();
    for (int s = 128; s > 0; s >>= 1) {
        if (t < s) red[t] += red[t + s];
        __syncthreads();
    }
    if (t == 0) {
        float z = red[0] + br[0];
        out[i] = 1.f / (1.f + expf(-z));
    }
}

// ---------------------------------------------------------------------------
extern "C" void kernel_launch(void* const* d_in, const int* in_sizes, int n_in,
                              void* d_out, int out_size, void* d_ws, size_t ws_size,
                              hipStream_t stream)
{
    const float* feature_obj  = (const float*)d_in[0];
    const float* highest_prob = (const float*)d_in[1];
    const float* rois         = (const float*)d_in[2];
    const float* emb          = (const float*)d_in[3];
    const float* W_vis        = (const float*)d_in[4];
    const float* b_vis        = (const float*)d_in[5];
    const float* Wk           = (const float*)d_in[6];
    const float* bk           = (const float*)d_in[7];
    const float* Wq           = (const float*)d_in[8];
    const float* bq           = (const float*)d_in[9];
    const float* Wv           = (const float*)d_in[10];
    const float* bv           = (const float*)d_in[11];
    const float* Wg           = (const float*)d_in[12];
    const float* bg           = (const float*)d_in[13];
    const float* Wr           = (const float*)d_in[14];
    const float* br           = (const float*)d_in[15];

    char* ws = (char*)d_ws;
    int*   rank = (int*)(ws);                                   // 256 ints
    float* x    = (float*)(ws + 1024);                          // 256x1024
    float* karr = x    + (size_t)N_OBJ * DIM_MM;                // 256x512
    float* qarr = karr + (size_t)N_OBJ * DIM_KEY;               // 256x512
    float* varr = qarr + (size_t)N_OBJ * DIM_KEY;               // 256x1024
    float* vw   = varr + (size_t)N_OBJ * DIM_MM;                // 256x256
    float* gw   = vw   + (size_t)N_OBJ * N_OBJ;                 // 256x256
    float* att  = gw   + (size_t)N_OBJ * N_OBJ;                 // 256x256
    float* feat = att  + (size_t)N_OBJ * N_OBJ;                 // 256x1024

    const float inv_sqrt_key = 0.044194173824159216f;  // 1/sqrt(512)

    rank_kernel<<<1, 256, 0, stream>>>(highest_prob, rank);

    // x = relu(emb[rank] + feature_obj @ W_vis^T + b_vis)
    gemm_wmma_f32<<<dim3(DIM_MM / BN, N_OBJ / BM), 256, 0, stream>>>(
        feature_obj, W_vis, x, b_vis, emb, rank,
        N_OBJ, DIM_MM, DIM_HO, DIM_HO, 1, 1.0f, 1);

    // k = x @ Wk^T + bk ; q = x @ Wq^T + bq ; v = x @ Wv^T + bv
    gemm_wmma_f32<<<dim3(DIM_KEY / BN, N_OBJ / BM), 256, 0, stream>>>(
        x, Wk, karr, bk, nullptr, nullptr,
        N_OBJ, DIM_KEY, DIM_MM, DIM_MM, 1, 1.0f, 0);
    gemm_wmma_f32<<<dim3(DIM_KEY / BN, N_OBJ / BM), 256, 0, stream>>>(
        x, Wq, qarr, bq, nullptr, nullptr,
        N_OBJ, DIM_KEY, DIM_MM, DIM_MM, 1, 1.0f, 0);
    gemm_wmma_f32<<<dim3(DIM_MM / BN, N_OBJ / BM), 256, 0, stream>>>(
        x, Wv, varr, bv, nullptr, nullptr,
        N_OBJ, DIM_MM, DIM_MM, DIM_MM, 1, 1.0f, 0);

    // vw = (k @ q^T) / sqrt(512)
    gemm_wmma_f32<<<dim3(N_OBJ / BN, N_OBJ / BM), 256, 0, stream>>>(
        karr, qarr, vw, nullptr, nullptr, nullptr,
        N_OBJ, N_OBJ, DIM_KEY, DIM_KEY, 1, inv_sqrt_key, 0);

    geometry_kernel<<<N_OBJ, 256, 0, stream>>>(rois, Wg, bg, gw);
    att_kernel<<<N_OBJ, 256, 0, stream>>>(vw, gw, att);

    // feat = relu(att @ v)   (B accessed non-transposed: ldbn=1, ldbk=DIM_MM)
    gemm_wmma_f32<<<dim3(DIM_MM / BN, N_OBJ / BM), 256, 0, stream>>>(
        att, varr, feat, nullptr, nullptr, nullptr,
        N_OBJ, DIM_MM, N_OBJ, 1, DIM_MM, 1.0f, 1);

    final_kernel<<<N_OBJ, 256, 0, stream>>>(feat, Wr, br, (float*)d_out);
}